// LinkPredictor_61280593379642
// MI455X (gfx1250) — compile-verified
//
#include <hip/hip_runtime.h>
#include <hip/hip_bf16.h>

typedef __attribute__((ext_vector_type(2))) float v2f;
typedef __attribute__((ext_vector_type(8))) float v8f;

#define TPB 256

// ---------------- utility ----------------
__global__ void zero_f32_kernel(float* __restrict__ p, long long n) {
  long long i = (long long)blockIdx.x * blockDim.x + threadIdx.x;
  if (i < n) p[i] = 0.0f;
}

// ---------------- degree / dinv ----------------
__global__ void deg_kernel(const long long* __restrict__ ei, float* __restrict__ deg, int E) {
  int e = blockIdx.x * blockDim.x + threadIdx.x;
  if (e < E) atomicAdd(&deg[(int)ei[(long long)E + e]], 1.0f);
}

__global__ void dinv_kernel(float* __restrict__ d, int N) {
  int n = blockIdx.x * blockDim.x + threadIdx.x;
  if (n < N) d[n] = rsqrtf(d[n] + 1.0f);  // +1 self loop
}

// ---------------- W -> per-lane WMMA fragment layout ----------------
// fragment index i = ((ks*8 + tile)*32 + lane); lane<16: K=4ks+{0,1}, col=lane
// lane>=16: K=4ks+{2,3}, col=lane-16.  (A-side mirrors this: half selects K pair)
__global__ void wfrag_kernel(const float* __restrict__ W, float2* __restrict__ wf) {
  int i = blockIdx.x * blockDim.x + threadIdx.x;
  if (i >= 32 * 8 * 32) return;
  int lane = i & 31, t = (i >> 5) & 7, ks = i >> 8;
  int c = lane & 15, kp = (lane >> 4) * 2, kk = ks * 4;
  wf[i] = make_float2(W[(kk + kp) * 128 + t * 16 + c],
                      W[(kk + kp + 1) * 128 + t * 16 + c]);
}

// ---------------- f32 WMMA GEMM: C[N,128] = A[N,128] @ W[128,128] ----------------
__global__ void gemm_wmma_kernel(const float* __restrict__ A,
                                 const float2* __restrict__ wf,
                                 float* __restrict__ C, int N) {
  int wave = blockIdx.x * (blockDim.x >> 5) + (threadIdx.x >> 5);
  int lane = threadIdx.x & 31;
  int row0 = wave << 4;
  if (row0 >= N) return;
  int half = lane >> 4;   // selects K pair {0,1} vs {2,3}
  int l15  = lane & 15;   // A row within strip / C column within tile
  int kp   = half << 1;

  int arow = row0 + l15;
  if (arow > N - 1) arow = N - 1;            // clamp for (unused) tail rows
  const float* ap = A + (size_t)arow * 128 + kp;

  v8f acc[8];
#pragma unroll
  for (int t = 0; t < 8; ++t) {
    v8f z = {0.f, 0.f, 0.f, 0.f, 0.f, 0.f, 0.f, 0.f};
    acc[t] = z;
  }

#pragma unroll 4
  for (int ks = 0; ks < 32; ++ks) {
    float2 av = *(const float2*)(ap + ks * 4);
    v2f a; a.x = av.x; a.y = av.y;
    const float2* wp = wf + ks * 256 + lane;
#pragma unroll
    for (int t = 0; t < 8; ++t) {
      float2 bv = wp[t * 32];
      v2f b; b.x = bv.x; b.y = bv.y;
      acc[t] = __builtin_amdgcn_wmma_f32_16x16x4_f32(
          false, a, false, b, (short)0, acc[t], false, false);
    }
  }

  // C/D layout: VGPR r -> M = r + 8*half ; column = l15
#pragma unroll
  for (int t = 0; t < 8; ++t) {
#pragma unroll
    for (int r = 0; r < 8; ++r) {
      int rr = row0 + (half << 3) + r;
      if (rr < N) C[(size_t)rr * 128 + t * 16 + l15] = acc[t][r];
    }
  }
}

// ---------------- agg = h * self_w + bias ----------------
__global__ void agg_init_kernel(const float* __restrict__ h, const float* __restrict__ dinv,
                                const float* __restrict__ b, float* __restrict__ agg, int N) {
  long long i = (long long)blockIdx.x * blockDim.x + threadIdx.x;
  int n = (int)(i >> 5);
  if (n >= N) return;
  int f = ((int)i & 31) << 2;
  float sw = dinv[n]; sw *= sw;
  float4 v  = *(const float4*)(h + (size_t)n * 128 + f);
  float4 bv = *(const float4*)(b + f);
  float4 o = make_float4(fmaf(v.x, sw, bv.x), fmaf(v.y, sw, bv.y),
                         fmaf(v.z, sw, bv.z), fmaf(v.w, sw, bv.w));
  *(float4*)(agg + (size_t)n * 128 + f) = o;
}

// ---------------- edge scatter: agg[dst] += w * h[src] ----------------
__global__ void scatter_kernel(const float* __restrict__ h, const long long* __restrict__ ei,
                               const float* __restrict__ dinv, float* __restrict__ agg, int E) {
  long long i = (long long)blockIdx.x * blockDim.x + threadIdx.x;
  int e = (int)(i >> 5);
  if (e >= E) return;
  int f = ((int)i & 31) << 2;
  int s = (int)ei[e];
  int d = (int)ei[(long long)E + e];
  float w = dinv[s] * dinv[d];
  float4 v = *(const float4*)(h + (size_t)s * 128 + f);
  float* o = agg + (size_t)d * 128 + f;
  atomicAdd(o + 0, w * v.x);
  atomicAdd(o + 1, w * v.y);
  atomicAdd(o + 2, w * v.z);
  atomicAdd(o + 3, w * v.w);
}

// ---------------- BatchNorm statistics ----------------
__global__ void bn_stats_kernel(const float* __restrict__ h, float* __restrict__ gsum,
                                float* __restrict__ gsq, int N) {
  __shared__ float s1[256], s2[256];
  int c  = threadIdx.x & 127;
  int r0 = threadIdx.x >> 7;
  int base = blockIdx.x * 256;
  int rend = base + 256; if (rend > N) rend = N;
  float s = 0.f, q = 0.f;
  for (int r = base + r0; r < rend; r += 2) {
    float v = h[(size_t)r * 128 + c];
    s += v; q += v * v;
  }
  s1[threadIdx.x] = s; s2[threadIdx.x] = q;
  __syncthreads();
  if (threadIdx.x < 128) {
    atomicAdd(&gsum[c], s1[c] + s1[c + 128]);
    atomicAdd(&gsq[c],  s2[c] + s2[c + 128]);
  }
}

__global__ void bn_final_kernel(const float* __restrict__ gsum, const float* __restrict__ gsq,
                                const float* __restrict__ g, const float* __restrict__ be,
                                float* __restrict__ sc, float* __restrict__ sh, float invN) {
  int c = threadIdx.x;
  if (c >= 128) return;
  float mu  = gsum[c] * invN;
  float var = gsq[c] * invN - mu * mu;
  float s = g[c] * rsqrtf(var + 1e-5f);
  sc[c] = s;
  sh[c] = be[c] - mu * s;
}

__global__ void bn_apply_kernel(float* __restrict__ h, const float* __restrict__ sc,
                                const float* __restrict__ sh, int N, int relu) {
  long long i = (long long)blockIdx.x * blockDim.x + threadIdx.x;
  int n = (int)(i >> 5);
  if (n >= N) return;
  int f = ((int)i & 31) << 2;
  float4 v = *(float4*)(h + (size_t)n * 128 + f);
  float4 a = *(const float4*)(sc + f);
  float4 b = *(const float4*)(sh + f);
  v.x = fmaf(v.x, a.x, b.x);
  v.y = fmaf(v.y, a.y, b.y);
  v.z = fmaf(v.z, a.z, b.z);
  v.w = fmaf(v.w, a.w, b.w);
  if (relu) {
    v.x = fmaxf(v.x, 0.f); v.y = fmaxf(v.y, 0.f);
    v.z = fmaxf(v.z, 0.f); v.w = fmaxf(v.w, 0.f);
  }
  *(float4*)(h + (size_t)n * 128 + f) = v;
}

// ---------------- link-prediction dot products ----------------
__global__ void dot_kernel(const float* __restrict__ h, const long long* __restrict__ eli,
                           float* __restrict__ out, int Q) {
  long long i = (long long)blockIdx.x * blockDim.x + threadIdx.x;
  int q = (int)(i >> 5);
  if (q >= Q) return;
  int lane = (int)i & 31;
  int u = (int)eli[q];
  int v = (int)eli[(long long)Q + q];
  float4 a = *(const float4*)(h + (size_t)u * 128 + lane * 4);
  float4 b = *(const float4*)(h + (size_t)v * 128 + lane * 4);
  float s = a.x * b.x + a.y * b.y + a.z * b.z + a.w * b.w;
#pragma unroll
  for (int off = 16; off > 0; off >>= 1) s += __shfl_down(s, off, 32);
  if (lane == 0) out[q] = s;
}

// ---------------- host ----------------
static inline int cdiv_i(long long a, long long b) { return (int)((a + b - 1) / b); }

extern "C" void kernel_launch(void* const* d_in, const int* in_sizes, int n_in,
                              void* d_out, int out_size, void* d_ws, size_t ws_size,
                              hipStream_t stream) {
  const float* x      = (const float*)d_in[0];
  const float* Wm[3]  = {(const float*)d_in[1], (const float*)d_in[5],  (const float*)d_in[9]};
  const float* bb[3]  = {(const float*)d_in[2], (const float*)d_in[6],  (const float*)d_in[10]};
  const float* gg[3]  = {(const float*)d_in[3], (const float*)d_in[7],  (const float*)d_in[11]};
  const float* be[3]  = {(const float*)d_in[4], (const float*)d_in[8],  (const float*)d_in[12]};
  const long long* ei  = (const long long*)d_in[13];
  const long long* eli = (const long long*)d_in[14];

  int N = in_sizes[0] / 128;
  int E = in_sizes[13] / 2;
  int Q = in_sizes[14] / 2;
  float* out = (float*)d_out;

  char* p = (char*)d_ws;
  float*  hA    = (float*)p;  p += (size_t)N * 128 * sizeof(float);
  float*  hB    = (float*)p;  p += (size_t)N * 128 * sizeof(float);
  float*  dinv  = (float*)p;  p += (((size_t)N * sizeof(float)) + 255) & ~(size_t)255;
  float2* wfrag = (float2*)p; p += (size_t)32 * 8 * 32 * sizeof(float2);
  float*  gsum  = (float*)p;  p += 128 * sizeof(float);
  float*  gsq   = (float*)p;  p += 128 * sizeof(float);
  float*  bnsc  = (float*)p;  p += 128 * sizeof(float);
  float*  bnsh  = (float*)p;  p += 128 * sizeof(float);

  // degrees -> dinv
  zero_f32_kernel<<<cdiv_i(N, TPB), TPB, 0, stream>>>(dinv, N);
  deg_kernel<<<cdiv_i(E, TPB), TPB, 0, stream>>>(ei, dinv, E);
  dinv_kernel<<<cdiv_i(N, TPB), TPB, 0, stream>>>(dinv, N);

  int strips = cdiv_i(N, 16);
  const float* hin = x;
  for (int L = 0; L < 3; ++L) {
    wfrag_kernel<<<cdiv_i(32 * 8 * 32, TPB), TPB, 0, stream>>>(Wm[L], wfrag);
    gemm_wmma_kernel<<<cdiv_i(strips, TPB / 32), TPB, 0, stream>>>(hin, wfrag, hA, N);
    agg_init_kernel<<<cdiv_i((long long)N * 32, TPB), TPB, 0, stream>>>(hA, dinv, bb[L], hB, N);
    scatter_kernel<<<cdiv_i((long long)E * 32, TPB), TPB, 0, stream>>>(hA, ei, dinv, hB, E);
    zero_f32_kernel<<<1, TPB, 0, stream>>>(gsum, 256);  // gsum & gsq are contiguous
    bn_stats_kernel<<<cdiv_i(N, 256), TPB, 0, stream>>>(hB, gsum, gsq, N);
    bn_final_kernel<<<1, 128, 0, stream>>>(gsum, gsq, gg[L], be[L], bnsc, bnsh, 1.0f / (float)N);
    bn_apply_kernel<<<cdiv_i((long long)N * 32, TPB), TPB, 0, stream>>>(hB, bnsc, bnsh, N, (L < 2) ? 1 : 0);
    hin = hB;
  }

  dot_kernel<<<cdiv_i((long long)Q * 32, TPB), TPB, 0, stream>>>(hB, eli, out, Q);
}